// FastKDTree_37744172597260
// MI455X (gfx1250) — compile-verified
//
#include <hip/hip_runtime.h>

typedef __attribute__((ext_vector_type(2))) float v2f;
typedef __attribute__((ext_vector_type(8))) float v8f;

#define D_DIM 256
#define WAVES 8
#define TILES 16
#define ROWS_PER_BLOCK (WAVES * 16 * TILES)  // 2048

// ---------------------------------------------------------------------------
// Pass 1: scores[b,n] = ||f_n||^2 - 2 * <q_b, f_n>   (monotone in distance)
// Dot products via V_WMMA_F32_16X16X4_F32: A = 16 feature rows x 4 dims,
// B = query replicated across all 16 N-columns, C accumulates over 64 chunks.
// ---------------------------------------------------------------------------
__global__ __launch_bounds__(256) void knn_score_kernel(
    const float* __restrict__ queries,
    const float* __restrict__ features,
    float* __restrict__ scores,
    int N)
{
    __shared__ float q_lds[D_DIM];
    __shared__ float dot_lds[WAVES * 16];

    const int b    = blockIdx.y;
    const int tid  = threadIdx.x;
    const int wave = tid >> 5;
    const int lane = tid & 31;
    const int half = lane >> 4;   // 0: lanes 0-15, 1: lanes 16-31
    const int lrow = lane & 15;

    // block-cooperative load of the query into LDS (256 floats)
    q_lds[tid] = queries[b * D_DIM + tid];
    __syncthreads();

    for (int t = 0; t < TILES; ++t) {
        const int rowbase = blockIdx.x * ROWS_PER_BLOCK + t * (WAVES * 16) + wave * 16;

        // lane L streams row (L%16); K-offset 2*half selects its A-fragment slice
        const float* fp = features + ((size_t)b * N + rowbase + lrow) * D_DIM + 2 * half;
        const float* qp = q_lds + 2 * half;

        v8f  c   = {0.f, 0.f, 0.f, 0.f, 0.f, 0.f, 0.f, 0.f};
        float nrm = 0.f;

        #pragma unroll 4
        for (int i = 0; i < D_DIM / 4; ++i) {
            v2f a  = *(const v2f*)(fp + 4 * i);   // f[row, k0+2h], f[row, k0+2h+1]
            v2f bq = *(const v2f*)(qp + 4 * i);   // q[k0+2h], q[k0+2h+1] (replicated cols)
            nrm += a.x * a.x + a.y * a.y;
            // D = A(16x4) * B(4x16) + C ; emits v_wmma_f32_16x16x4_f32
            c = __builtin_amdgcn_wmma_f32_16x16x4_f32(
                    false, a, false, bq, (short)0, c, false, false);
        }

        // full row norm: lane L and L^16 hold the two K-halves of row L%16
        float nfull = nrm + __shfl_xor(nrm, 16, 32);

        // C layout: lanes 0-15 hold rows 0..7 in c[0..7]; lanes 16-31 hold rows 8..15.
        // All columns identical (query replicated) -> lane 0 / lane 16 publish.
        if (lane == 0) {
            #pragma unroll
            for (int v = 0; v < 8; ++v) dot_lds[wave * 16 + v] = c[v];
        }
        if (lane == 16) {
            #pragma unroll
            for (int v = 0; v < 8; ++v) dot_lds[wave * 16 + 8 + v] = c[v];
        }
        __syncthreads();

        if (lane < 16) {
            float dot = dot_lds[wave * 16 + lane];
            scores[(size_t)b * N + rowbase + lane] = nfull - 2.0f * dot;
        }
        __syncthreads();
    }
}

// ---------------------------------------------------------------------------
// Pass 2: per batch, select 16 smallest scores (lowest-index tie-break, like
// top_k stability) and gather the winning feature rows to the output.
// One 256-thread block per batch.
// ---------------------------------------------------------------------------
__global__ __launch_bounds__(256) void topk_gather_kernel(
    const float* __restrict__ features,
    const float* __restrict__ scores,
    float* __restrict__ out,
    int N)
{
    const int b   = blockIdx.x;
    const int tid = threadIdx.x;
    const float INF = __builtin_inff();

    // per-thread sorted top-16 (ascending)
    float best[16];
    int   bidx[16];
    #pragma unroll
    for (int v = 0; v < 16; ++v) { best[v] = INF; bidx[v] = 0x7fffffff; }

    for (int n = tid; n < N; n += 256) {
        float s = scores[(size_t)b * N + n];
        if (s < best[15]) {
            int v = 15;
            while (v > 0 && best[v - 1] > s) {
                best[v] = best[v - 1]; bidx[v] = bidx[v - 1]; --v;
            }
            best[v] = s; bidx[v] = n;
        }
    }

    __shared__ float cs[256 * 16];
    __shared__ int   ci[256 * 16];
    __shared__ float rv[256];
    __shared__ int   rp[256];
    __shared__ int   sel[16];

    #pragma unroll
    for (int v = 0; v < 16; ++v) {
        cs[tid * 16 + v] = best[v];
        ci[tid * 16 + v] = bidx[v];
    }
    __syncthreads();

    for (int k = 0; k < 16; ++k) {
        // local argmin over this thread's 16 candidate slots
        float mv = INF; int mp = tid * 16; int mi = 0x7fffffff;
        #pragma unroll
        for (int v = 0; v < 16; ++v) {
            int p = tid * 16 + v;
            float s = cs[p];
            int   i = ci[p];
            if (s < mv || (s == mv && i < mi)) { mv = s; mp = p; mi = i; }
        }
        rv[tid] = mv; rp[tid] = mp;
        __syncthreads();

        // block tree-reduction (tie-break on feature index)
        for (int off = 128; off > 0; off >>= 1) {
            if (tid < off) {
                float s1 = rv[tid];       int p1 = rp[tid];
                float s2 = rv[tid + off]; int p2 = rp[tid + off];
                bool take = (s2 < s1) || (s2 == s1 && ci[p2] < ci[p1]);
                if (take) { rv[tid] = s2; rp[tid] = p2; }
            }
            __syncthreads();
        }

        if (tid == 0) {
            int p = rp[0];
            sel[k] = ci[p];
            cs[p]  = INF;   // remove winner
        }
        __syncthreads();
    }

    // gather: out[b,k,:] = features[b, sel[k], :]   (coalesced)
    for (int e = tid; e < 16 * D_DIM; e += 256) {
        int k = e >> 8;
        int d = e & (D_DIM - 1);
        out[((size_t)b * 16 + k) * D_DIM + d] =
            features[((size_t)b * N + sel[k]) * D_DIM + d];
    }
}

extern "C" void kernel_launch(void* const* d_in, const int* in_sizes, int n_in,
                              void* d_out, int out_size, void* d_ws, size_t ws_size,
                              hipStream_t stream) {
    (void)n_in; (void)out_size; (void)ws_size;
    const float* queries  = (const float*)d_in[0];   // [B, 256]
    const float* features = (const float*)d_in[1];   // [B, N, 256]
    float*       out      = (float*)d_out;           // [B, 16, 256]

    const int D = D_DIM;
    const int B = in_sizes[0] / D;                       // 32
    const long long totalF = (long long)in_sizes[1];
    const int N = (int)(totalF / ((long long)B * D));    // 65536

    float* scores = (float*)d_ws;                        // B*N floats (8 MB)

    dim3 grid1(N / ROWS_PER_BLOCK, B);
    knn_score_kernel<<<grid1, 256, 0, stream>>>(queries, features, scores, N);
    topk_gather_kernel<<<B, 256, 0, stream>>>(features, scores, out, N);
}